// MoCo_14688788152456
// MI455X (gfx1250) — compile-verified
//
#include <hip/hip_runtime.h>
#include <hip/hip_bf16.h>

typedef __attribute__((ext_vector_type(2))) float v2f;
typedef __attribute__((ext_vector_type(4))) float v4f;
typedef __attribute__((ext_vector_type(8))) float v8f;

#define N_SAMP 512
#define C_DIM  128
#define K_DIM  65536
#define T_CLS  7
#define INV_TEMP (1.0f / 0.07f)

// d_out layout (floats/ints, flat, reference return order):
//   logits  [512 x 65537]            @ 0
//   labels  [512] (int zeros)        @ 33554944
//   new_queues [7 x 128 x 65536]     @ 33555456
//   new_ptrs [7] (int)               @ 92275712
#define OFF_LABELS   33554944UL
#define OFF_NEWQ     33555456UL
#define OFF_NEWPTRS  92275712UL

// ws (ints): [0..7] cls_offset prefix, [8..519] idx_list (class-grouped sample ids),
//            [520..1031] pos (enqueue slot per sample)

// ---------------------------------------------------------------------------
// Stage: l_pos, labels, stable class grouping, new_ptrs
// ---------------------------------------------------------------------------
__global__ __launch_bounds__(512) void moco_stage(
    const float* __restrict__ q, const float* __restrict__ kk,
    const int* __restrict__ targets, const int* __restrict__ ptrs,
    float* __restrict__ out, int* __restrict__ ws) {
  const int n = threadIdx.x;  // one block of 512
  // l_pos = q[n] . k[n]
  const v4f* qa = reinterpret_cast<const v4f*>(q + n * C_DIM);
  const v4f* ka = reinterpret_cast<const v4f*>(kk + n * C_DIM);
  float s = 0.f;
  #pragma unroll
  for (int c = 0; c < C_DIM / 4; ++c) {
    v4f a = qa[c], b = ka[c];
    s += a.x * b.x + a.y * b.y + a.z * b.z + a.w * b.w;
  }
  out[(size_t)n * (K_DIM + 1)] = s * INV_TEMP;
  reinterpret_cast<int*>(out + OFF_LABELS)[n] = 0;

  if (n == 0) {
    int cnt[T_CLS];
    #pragma unroll
    for (int t = 0; t < T_CLS; ++t) cnt[t] = 0;
    for (int i = 0; i < N_SAMP; ++i) cnt[targets[i]]++;
    int off[T_CLS + 1];
    off[0] = 0;
    #pragma unroll
    for (int t = 0; t < T_CLS; ++t) off[t + 1] = off[t] + cnt[t];
    #pragma unroll
    for (int t = 0; t < T_CLS + 1; ++t) ws[t] = off[t];
    int cur[T_CLS];
    #pragma unroll
    for (int t = 0; t < T_CLS; ++t) cur[t] = 0;
    for (int i = 0; i < N_SAMP; ++i) {           // stable: matches cumsum rank
      int t = targets[i];
      int r = cur[t]++;
      ws[8 + off[t] + r] = i;                    // idx_list
      ws[8 + N_SAMP + i] = (ptrs[t] + r) & (K_DIM - 1);  // pos
    }
    int* new_ptrs = reinterpret_cast<int*>(out + OFF_NEWPTRS);
    #pragma unroll
    for (int t = 0; t < T_CLS; ++t)
      new_ptrs[t] = (ptrs[t] + cnt[t]) & (K_DIM - 1);
  }
}

// ---------------------------------------------------------------------------
// Scatter enqueued k columns into new_queues (runs AFTER gemm's fused copy)
// ---------------------------------------------------------------------------
__global__ __launch_bounds__(128) void moco_scatter(
    const float* __restrict__ kk, const int* __restrict__ targets,
    const int* __restrict__ ws, float* __restrict__ newq) {
  const int n = blockIdx.x;     // 512 blocks
  const int c = threadIdx.x;    // 128 threads
  const int t = targets[n];
  const int p = ws[8 + N_SAMP + n];
  newq[((size_t)t * C_DIM + c) * K_DIM + p] = kk[n * C_DIM + c];
}

// ---------------------------------------------------------------------------
// Class-grouped GEMM via V_WMMA_F32_16X16X4_F32, with the new_queues bulk
// copy FUSED into the B-tile staging (queues are read from HBM exactly once).
// grid = (K/128, 7); block = 256 (8 waves). Wave w owns 16 output columns.
// Streaming traffic (queue read, newq write, logits write) is non-temporal.
// ---------------------------------------------------------------------------
#define TN 128          // columns per workgroup
#define BS_STRIDE 136   // LDS pad: (2*136)%64 == 16 -> half-waves hit disjoint banks
#define AS_STRIDE 132

__global__ __launch_bounds__(256) void moco_gemm(
    const float* __restrict__ q, const float* __restrict__ queues,
    const int* __restrict__ ws, float* __restrict__ out,
    float* __restrict__ newq) {
  const int t = blockIdx.y;
  const int col0 = blockIdx.x * TN;
  const int tid = threadIdx.x;
  const int lane = tid & 31;
  const int wv = tid >> 5;        // 0..7
  const int half = lane >> 4;     // 0/1
  const int l16 = lane & 15;

  __shared__ float Bs[C_DIM][BS_STRIDE];
  __shared__ float As[16][AS_STRIDE];
  __shared__ int Arow[16];

  // Stage B tile (128 c-rows x 128 cols) AND emit the new_queues copy from
  // the same registers: one HBM read serves both GEMM and queue copy.
  const float* qt = queues + (size_t)t * C_DIM * K_DIM + col0;
  float* nq = newq + (size_t)t * C_DIM * K_DIM + col0;
  for (int i = tid; i < C_DIM * (TN / 4); i += 256) {
    const int c = i >> 5;           // 0..127
    const int v = (i & 31) * 4;     // 0..124
    const v4f* gp = reinterpret_cast<const v4f*>(qt + (size_t)c * K_DIM + v);
    v4f d = __builtin_nontemporal_load(gp);
    __builtin_nontemporal_store(d, reinterpret_cast<v4f*>(nq + (size_t)c * K_DIM + v));
    Bs[c][v + 0] = d.x; Bs[c][v + 1] = d.y; Bs[c][v + 2] = d.z; Bs[c][v + 3] = d.w;
  }

  const int cls_beg = ws[t];
  const int cnt = ws[t + 1] - cls_beg;
  if (cnt == 0) return;  // copy already emitted above

  const int ntiles = (cnt + 15) >> 4;
  for (int tile = 0; tile < ntiles; ++tile) {
    if (tid < 16) {
      const int row = tile * 16 + tid;
      Arow[tid] = (row < cnt) ? ws[8 + cls_beg + row] : -1;
    }
    __syncthreads();
    // Stage A tile: 16 gathered q rows x 128 (zero-pad invalid rows)
    for (int i = tid; i < 16 * (C_DIM / 4); i += 256) {
      const int r = i >> 5;
      const int v = (i & 31) * 4;
      const int smp = Arow[r];
      v4f d = {0.f, 0.f, 0.f, 0.f};
      if (smp >= 0) d = *reinterpret_cast<const v4f*>(q + (size_t)smp * C_DIM + v);
      As[r][v + 0] = d.x; As[r][v + 1] = d.y; As[r][v + 2] = d.z; As[r][v + 3] = d.w;
    }
    __syncthreads();

    // 16x16 output per wave, accumulate over C in K=4 WMMA steps.
    v8f acc = {};
    #pragma unroll
    for (int c = 0; c < C_DIM; c += 4) {
      v2f a, b;
      // A 16x4 layout: VGPR0 = {K=c | K=c+2}, VGPR1 = {K=c+1 | K=c+3}, M=l16
      a.x = As[l16][c + 2 * half];
      a.y = As[l16][c + 2 * half + 1];
      // B 4x16 layout: row-striped across lanes; halves hold K=c..c+1 / c+2..c+3
      b.x = Bs[c + 2 * half][wv * 16 + l16];
      b.y = Bs[c + 2 * half + 1][wv * 16 + l16];
      acc = __builtin_amdgcn_wmma_f32_16x16x4_f32(
          false, a, false, b, (short)0, acc, false, false);
    }

    // D layout: VGPR r -> M = r + 8*half, N = l16
    #pragma unroll
    for (int r = 0; r < 8; ++r) {
      const int m = r + half * 8;
      const int row = tile * 16 + m;
      if (row < cnt) {
        const int smp = Arow[m];
        __builtin_nontemporal_store(
            acc[r] * INV_TEMP,
            out + (size_t)smp * (K_DIM + 1) + 1 + col0 + wv * 16 + l16);
      }
    }
    __syncthreads();  // before next tile overwrites As/Arow
  }
}

// ---------------------------------------------------------------------------
extern "C" void kernel_launch(void* const* d_in, const int* in_sizes, int n_in,
                              void* d_out, int out_size, void* d_ws, size_t ws_size,
                              hipStream_t stream) {
  const float* q       = (const float*)d_in[0];
  const float* kk      = (const float*)d_in[1];
  const int*   targets = (const int*)d_in[2];
  const float* queues  = (const float*)d_in[3];
  const int*   ptrs    = (const int*)d_in[4];
  float* out = (float*)d_out;
  int*   ws  = (int*)d_ws;

  // 1) l_pos, labels, class grouping, new_ptrs
  moco_stage<<<1, 512, 0, stream>>>(q, kk, targets, ptrs, out, ws);

  // 2) class-grouped WMMA GEMM for l_neg + fused queues->new_queues copy
  dim3 grid(K_DIM / TN, T_CLS);
  moco_gemm<<<grid, 256, 0, stream>>>(q, queues, ws, out, out + OFF_NEWQ);

  // 3) scatter enqueued k columns over the fused copy
  moco_scatter<<<N_SAMP, 128, 0, stream>>>(kk, targets, ws, out + OFF_NEWQ);
}